// Another_P_at_K_Loss_55817394979143
// MI455X (gfx1250) — compile-verified
//
#include <hip/hip_runtime.h>
#include <math.h>

typedef float v2f __attribute__((ext_vector_type(2)));
typedef float v8f __attribute__((ext_vector_type(8)));

#define BSZ 4096
#define DSZ 512
#define NEGV (-1e30f)
// k * tau * log(THRESH) = 4 * 1 * log(100)
#define HARD_GAP 18.420680743952367f

__device__ __forceinline__ float lae(float a, float b) {
  float m = fmaxf(a, b);
  float d = fabsf(a - b);
  return m + log1pf(__expf(-d));
}

// Async global->LDS copy (GVS mode: 64-bit SGPR base + 32-bit VGPR byte offset),
// 16 bytes per lane, tracked by ASYNCcnt. VDST = LDS byte address (low 32 bits of
// the generic LDS pointer).
__device__ __forceinline__ void async_ld128(unsigned lds_addr, unsigned gbyte_off,
                                            const float* base) {
  asm volatile("global_load_async_to_lds_b128 %0, %1, %2"
               :
               : "v"(lds_addr), "v"(gbyte_off), "s"(base)
               : "memory");
}

// ---------------- accumulator init ----------------
__global__ void init_kernel(float* __restrict__ accs) {
  accs[0] = 0.0f;  // sum of (F_full - F_pos)
  accs[1] = 0.0f;  // sum of picked positives
  accs[2] = 0.0f;  // sum of squares of (cov - I)
  accs[3] = 0.0f;
}

// ---------------- column means of E ----------------
__global__ __launch_bounds__(DSZ) void col_mean_kernel(const float* __restrict__ E,
                                                       float* __restrict__ mean) {
  const int c = threadIdx.x;
  float s = 0.0f;
  for (int b = 0; b < BSZ; ++b) s += E[(size_t)b * DSZ + c];
  mean[c] = s * (1.0f / (float)BSZ);
}

// ---------------- score_hat = E @ E^T (+margin, diag=-inf) ----------------
// 64x64 output tile per block; 16 waves, each one 16x16 WMMA tile.
// K chunked by 64 via double-buffered LDS filled with async global->LDS b128.
__global__ __launch_bounds__(512) void score_gemm(const float* __restrict__ E,
                                                  const int* __restrict__ lab,
                                                  float* __restrict__ S) {
  __shared__ float As[2][64][68];  // [buf][row-local][k-local]
  __shared__ float Bs[2][64][68];  // [buf][col-local][k-local]
  const int i0 = blockIdx.y * 64;
  const int j0 = blockIdx.x * 64;
  const int tid = threadIdx.x;
  const int lane = tid & 31;
  const int wave = tid >> 5;
  const int tm = (wave >> 2) * 16;
  const int tn = (wave & 3) * 16;
  const int mrow = lane & 15;      // M (or N) index for A/B operands
  const int kb = (lane >> 4) * 2;  // K base within 4-wide step

  // Issue the async fills for one 64-wide K chunk: 4 b128 per thread (2 for A, 2 for B).
  auto issue = [&](int buf, int k0) {
#pragma unroll
    for (int t = 0; t < 2; ++t) {
      int idx4 = t * 512 + tid;      // 0..1023 vec4 slots
      int r = idx4 >> 4;             // 0..63
      int c = (idx4 & 15) * 4;       // 0..60 step 4
      unsigned la = (unsigned)(uintptr_t)&As[buf][r][c];
      unsigned lb = (unsigned)(uintptr_t)&Bs[buf][r][c];
      unsigned ga = (unsigned)(((i0 + r) * DSZ + k0 + c) * 4);
      unsigned gb = (unsigned)(((j0 + r) * DSZ + k0 + c) * 4);
      async_ld128(la, ga, E);
      async_ld128(lb, gb, E);
    }
  };

  v8f acc = {};
  issue(0, 0);
  for (int chunk = 0; chunk < DSZ / 64; ++chunk) {
    const int buf = chunk & 1;
    if (chunk + 1 < DSZ / 64) {
      issue(buf ^ 1, (chunk + 1) * 64);
      asm volatile("s_wait_asynccnt 0x4" ::: "memory");  // 4 newest = next chunk
    } else {
      asm volatile("s_wait_asynccnt 0x0" ::: "memory");
    }
    __syncthreads();  // all waves' fills for this chunk have landed
#pragma unroll
    for (int kk = 0; kk < 64; kk += 4) {
      v2f a, b;
      a.x = As[buf][tm + mrow][kk + kb];
      a.y = As[buf][tm + mrow][kk + kb + 1];
      b.x = Bs[buf][tn + mrow][kk + kb];
      b.y = Bs[buf][tn + mrow][kk + kb + 1];
      acc = __builtin_amdgcn_wmma_f32_16x16x4_f32(false, a, false, b, (short)0, acc,
                                                  false, false);
    }
    __syncthreads();  // buffer free for the fill two chunks ahead
  }

  // Epilogue: diagonal -> NEG, different-label -> +MARGIN
  const int n = lane & 15;
  const int mb = (lane >> 4) * 8;
  const int gj = j0 + tn + n;
  const int lj = lab[gj];
#pragma unroll
  for (int v = 0; v < 8; ++v) {
    int gi = i0 + tm + mb + v;
    float val = acc[v];
    if (gi == gj) val = NEGV;
    else if (lab[gi] != lj) val += 0.2f;
    S[(size_t)gi * BSZ + gj] = val;
  }
}

// ---------------- cov = ec^T ec / B ; accumulate ||cov - I||_F^2 ----------------
__global__ __launch_bounds__(512) void cov_gemm(const float* __restrict__ E,
                                                const float* __restrict__ mean,
                                                float* __restrict__ acc_cov) {
  __shared__ float As[64][68];  // [i-local][k-local] (transposed on load)
  __shared__ float Bs[64][68];  // [k-local][j-local]
  const int i0 = blockIdx.y * 64;
  const int j0 = blockIdx.x * 64;
  const int tid = threadIdx.x;
  const int lane = tid & 31;
  const int wave = tid >> 5;
  const int tm = (wave >> 2) * 16;
  const int tn = (wave & 3) * 16;
  const int mrow = lane & 15;
  const int kb = (lane >> 4) * 2;

  v8f acc = {};
  for (int k0 = 0; k0 < BSZ; k0 += 64) {
#pragma unroll
    for (int t = 0; t < 8; ++t) {
      int idx = t * 512 + tid;
      int r = idx >> 6;   // k-local (batch row)
      int c = idx & 63;   // feature-local
      As[c][r] = E[(size_t)(k0 + r) * DSZ + i0 + c] - mean[i0 + c];
      Bs[r][c] = E[(size_t)(k0 + r) * DSZ + j0 + c] - mean[j0 + c];
    }
    __syncthreads();
#pragma unroll
    for (int kk = 0; kk < 64; kk += 4) {
      v2f a, b;
      a.x = As[tm + mrow][kk + kb];
      a.y = As[tm + mrow][kk + kb + 1];
      b.x = Bs[kk + kb][tn + mrow];
      b.y = Bs[kk + kb + 1][tn + mrow];
      acc = __builtin_amdgcn_wmma_f32_16x16x4_f32(false, a, false, b, (short)0, acc,
                                                  false, false);
    }
    __syncthreads();
  }

  const int n = lane & 15;
  const int mb = (lane >> 4) * 8;
  const int gj = j0 + tn + n;
  float part = 0.0f;
#pragma unroll
  for (int v = 0; v < 8; ++v) {
    int gi = i0 + tm + mb + v;
    float cv = acc[v] * (1.0f / (float)BSZ);
    float d = cv - ((gi == gj) ? 1.0f : 0.0f);
    part += d * d;
  }
#pragma unroll
  for (int off = 16; off > 0; off >>= 1) part += __shfl_down(part, off);
  if (lane == 0) atomicAdd(acc_cov, part);
}

// ---------------- per-row top-k + smooth top-k (log-ESP) + err_pos ----------------
__global__ __launch_bounds__(256) void row_kernel(const float* __restrict__ S,
                                                  float* __restrict__ accs) {
  __shared__ float tv[8][32][5];
  __shared__ int   tc[8][32][5];
  __shared__ float le_s[8][32][5];
  const int wave = threadIdx.x >> 5;
  const int lane = threadIdx.x & 31;
  const int row = blockIdx.x * 8 + wave;
  const float* Srow = S + (size_t)row * BSZ;

  float bv0 = NEGV, bv1 = NEGV, bv2 = NEGV, bv3 = NEGV, bv4 = NEGV;
  int   bc0 = -1, bc1 = -1, bc2 = -1, bc3 = -1, bc4 = -1;
  float L0 = 0.0f, L1 = NEGV, L2 = NEGV, L3 = NEGV, L4 = NEGV;

  for (int c = lane; c < BSZ; c += 32) {
    float v = Srow[c];
    float z = v * 0.25f;  // x / (k*tau)
    L4 = lae(L4, L3 + z);
    L3 = lae(L3, L2 + z);
    L2 = lae(L2, L1 + z);
    L1 = lae(L1, L0 + z);
    float nv = v; int nc = c;
    if (nv > bv0) { float t = bv0; bv0 = nv; nv = t; int ti = bc0; bc0 = nc; nc = ti; }
    if (nv > bv1) { float t = bv1; bv1 = nv; nv = t; int ti = bc1; bc1 = nc; nc = ti; }
    if (nv > bv2) { float t = bv2; bv2 = nv; nv = t; int ti = bc2; bc2 = nc; nc = ti; }
    if (nv > bv3) { float t = bv3; bv3 = nv; nv = t; int ti = bc3; bc3 = nc; nc = ti; }
    if (nv > bv4) { bv4 = nv; bc4 = nc; }
  }
  tv[wave][lane][0] = bv0; tv[wave][lane][1] = bv1; tv[wave][lane][2] = bv2;
  tv[wave][lane][3] = bv3; tv[wave][lane][4] = bv4;
  tc[wave][lane][0] = bc0; tc[wave][lane][1] = bc1; tc[wave][lane][2] = bc2;
  tc[wave][lane][3] = bc3; tc[wave][lane][4] = bc4;
  le_s[wave][lane][0] = L0; le_s[wave][lane][1] = L1; le_s[wave][lane][2] = L2;
  le_s[wave][lane][3] = L3; le_s[wave][lane][4] = L4;
  __syncthreads();

  if (lane == 0) {
    // merge top-5 across lanes
    float mv[5]; int mc[5];
#pragma unroll
    for (int s = 0; s < 5; ++s) { mv[s] = NEGV; mc[s] = -1; }
    for (int l = 0; l < 32; ++l) {
#pragma unroll
      for (int s = 0; s < 5; ++s) {
        float nv = tv[wave][l][s]; int nc = tc[wave][l][s];
#pragma unroll
        for (int q = 0; q < 5; ++q) {
          if (nv > mv[q]) { float t = mv[q]; mv[q] = nv; nv = t; int ti = mc[q]; mc[q] = nc; nc = ti; }
        }
      }
    }
    // merge ESP states (log-space convolution; order-independent)
    float A0 = le_s[wave][0][0], A1 = le_s[wave][0][1], A2 = le_s[wave][0][2],
          A3 = le_s[wave][0][3], A4 = le_s[wave][0][4];
    for (int l = 1; l < 32; ++l) {
      float B0 = le_s[wave][l][0], B1 = le_s[wave][l][1], B2 = le_s[wave][l][2],
            B3 = le_s[wave][l][3], B4 = le_s[wave][l][4];
      float o4 = lae(lae(lae(lae(A0 + B4, A1 + B3), A2 + B2), A3 + B1), A4 + B0);
      float o3 = lae(lae(lae(A0 + B3, A1 + B2), A2 + B1), A3 + B0);
      float o2 = lae(lae(A0 + B2, A1 + B1), A2 + B0);
      float o1 = lae(A0 + B1, A1 + B0);
      A1 = o1; A2 = o2; A3 = o3; A4 = o4;  // A0 stays 0
    }
    // positive set: the other 7 entries of this row's contiguous group of 8
    float pv[5];
#pragma unroll
    for (int s = 0; s < 5; ++s) pv[s] = NEGV;
    float P0 = 0.0f, P1 = NEGV, P2 = NEGV, P3 = NEGV, P4 = NEGV;
    int g0 = row & ~7;
    for (int t = 0; t < 8; ++t) {
      int c = g0 + t;
      if (c == row) continue;
      float v = Srow[c];
      float z = v * 0.25f;
      P4 = lae(P4, P3 + z); P3 = lae(P3, P2 + z); P2 = lae(P2, P1 + z); P1 = lae(P1, P0 + z);
      float nv = v;
#pragma unroll
      for (int q = 0; q < 5; ++q) {
        if (nv > pv[q]) { float tt = pv[q]; pv[q] = nv; nv = tt; }
      }
    }
    float f_full = (mv[3] - mv[4] >= HARD_GAP)
                       ? (mv[0] + mv[1] + mv[2] + mv[3]) * 0.25f : A4;
    float f_pos = (pv[3] - pv[4] >= HARD_GAP)
                      ? (pv[0] + pv[1] + pv[2] + pv[3]) * 0.25f : P4;
    int gr = row >> 3;
    float picked = 0.0f;
#pragma unroll
    for (int s = 0; s < 4; ++s) picked += ((mc[s] >> 3) == gr) ? 1.0f : 0.0f;
    atomicAdd(&accs[0], f_full - f_pos);
    atomicAdd(&accs[1], picked);
  }
}

// ---------------- finalize ----------------
__global__ void final_kernel(const float* __restrict__ accs, float* __restrict__ out) {
  float loss1 = accs[0] * (1.0f / (float)BSZ);
  float loss3 = sqrtf(accs[2]);
  out[0] = loss1 + 0.1f * loss3;
  out[1] = (float)(BSZ * 4) - accs[1];  // err_pos = B*min(K,pm1) - picked
}

extern "C" void kernel_launch(void* const* d_in, const int* in_sizes, int n_in,
                              void* d_out, int out_size, void* d_ws, size_t ws_size,
                              hipStream_t stream) {
  (void)in_sizes; (void)n_in; (void)out_size; (void)ws_size;
  const float* E = (const float*)d_in[0];
  const int* lab = (const int*)d_in[1];
  float* ws = (float*)d_ws;
  float* S = ws;                                  // 4096*4096 floats (64 MB)
  float* mean = ws + (size_t)BSZ * BSZ;           // 512 floats
  float* accs = mean + DSZ;                       // 4 floats

  hipLaunchKernelGGL(init_kernel, dim3(1), dim3(1), 0, stream, accs);
  hipLaunchKernelGGL(col_mean_kernel, dim3(1), dim3(DSZ), 0, stream, E, mean);
  hipLaunchKernelGGL(score_gemm, dim3(BSZ / 64, BSZ / 64), dim3(512), 0, stream, E, lab, S);
  hipLaunchKernelGGL(cov_gemm, dim3(DSZ / 64, DSZ / 64), dim3(512), 0, stream, E, mean, accs + 2);
  hipLaunchKernelGGL(row_kernel, dim3(BSZ / 8), dim3(256), 0, stream, S, accs);
  hipLaunchKernelGGL(final_kernel, dim3(1), dim3(1), 0, stream, accs, (float*)d_out);
}